// Net_29137058136132
// MI455X (gfx1250) — compile-verified
//
#include <hip/hip_runtime.h>

// ---------------------------------------------------------------------------
// CDNA5 (gfx1250) fp32 WMMA implementation of the 6-layer Rs_GCN stack.
// GEMMs run on V_WMMA_F32_16X16X4_F32 with LDS tiles staged by the Tensor
// Data Mover (tensor_load_to_lds, TENSORcnt-synchronized, double-buffered).
// ---------------------------------------------------------------------------

typedef float        v2f __attribute__((ext_vector_type(2)));
typedef float        v8f __attribute__((ext_vector_type(8)));
typedef unsigned int v4u __attribute__((ext_vector_type(4)));
typedef int          v8i __attribute__((ext_vector_type(8)));
typedef int          v4i __attribute__((ext_vector_type(4)));

#define BDIM   256   // batch
#define NPTS   100   // points per batch
#define NPAD   112   // 7 * 16
#define CCH    512   // channels
#define DIMG_K 2048  // encoder K

// WMMA f32 16x16x4: D = A(16x4) * B(4x16) + C(16x16)
__device__ __forceinline__ v8f wmma4(v2f a, v2f b, v8f c) {
  return __builtin_amdgcn_wmma_f32_16x16x4_f32(
      /*neg_a=*/false, a, /*neg_b=*/false, b,
      /*c_mod=*/(short)0, c, /*reuse_a=*/false, /*reuse_b=*/false);
}

// ---------------------------------------------------------------------------
// TDM: DMA a 2D tile (rows x 32 floats, row stride = K floats) from global
// into LDS, inserting 4 pad DWORDs after every 32 DWORDs stored
// (-> LDS row stride 36 floats). Wave-level op, tracked by TENSORcnt.
//   group1 w0: data_size=4B (2<<16) | pad_enable (1<<20)
//            | pad_interval=32dw (4<<22) | pad_amount=4dw (3<<25)
// ---------------------------------------------------------------------------
__device__ __forceinline__ void tdm_load_tile32(unsigned lds_addr,
                                                const float* gptr,
                                                int rows, int K) {
  unsigned long long ga = (unsigned long long)(uintptr_t)gptr;
  v4u g0;
  g0[0] = 1u;                                            // count=1, user desc
  g0[1] = lds_addr;                                      // LDS byte address
  g0[2] = (unsigned)ga;                                  // global addr lo
  g0[3] = (unsigned)((ga >> 32) & 0x01FFFFFFu) | (2u << 30);  // hi25 | type=2
  v8i g1;
  g1[0] = 0x07120000;          // flags (see above), workgroup_mask=0
  g1[1] = (32 << 16);          // tensor_dim0 = 32 (lo16 in [31:16])
  g1[2] = (rows << 16);        // tensor_dim0 hi16=0 | tensor_dim1 lo16
  g1[3] = (32 << 16);          // tensor_dim1 hi16=0 | tile_dim0 = 32
  g1[4] = rows;                // tile_dim1 = rows | tile_dim2 = 0
  g1[5] = K;                   // tensor_dim0_stride lo32 (floats)
  g1[6] = 0;                   // stride hi16 | tensor_dim1_stride lo16
  g1[7] = 0;
  v4i z4 = {0, 0, 0, 0};
  v8i z8 = {0, 0, 0, 0, 0, 0, 0, 0};
  __builtin_amdgcn_tensor_load_to_lds(g0, g1, z4, z4, z8, 0);
}

// ---------------------------------------------------------------------------
// Generic GEMM:  out[M,N] = A[M,K] @ W[N,K]^T + bias[N]   (mode 0)
// mode 1 adds fused BatchNorm(eval) + residual on xres.
// Block: 128 threads (4 waves). Block tile 128(M) x 64(N), K slab = 32.
// Wave w computes rows [w*32, w*32+32) x 64 cols -> 2x4 v8f accumulators.
// A tile and W tile both staged row-major (stride 36) via TDM, double-buffered.
// B-frag = two consecutive floats of a W row -> aligned ds_load_b64.
// ---------------------------------------------------------------------------
#define TM 128
#define TN 64
#define TK 32
#define SL 36   // padded LDS row stride (floats)

__global__ __launch_bounds__(128)
void gemm_wmma_f32(const float* __restrict__ A, const float* __restrict__ W,
                   const float* __restrict__ bias, float* __restrict__ out,
                   const float* __restrict__ xres,
                   const float* __restrict__ gamma, const float* __restrict__ var,
                   const float* __restrict__ mean, const float* __restrict__ beta,
                   int M, int N, int K, int mode) {
  __shared__ float lA[2][TM * SL];
  __shared__ float lB[2][TN * SL];

  const int tid   = threadIdx.x;
  const int lane  = tid & 31;
  const int wave  = tid >> 5;      // 0..3
  const int l16   = lane & 15;
  const int lhalf = lane >> 4;     // 0/1
  const int m0    = blockIdx.x * TM;
  const int n0    = blockIdx.y * TN;
  const int wrow  = wave * 32;

  v8f acc[2][4] = {};              // 2 row tiles x 4 col tiles

  const int nt = K / TK;
  if (wave == 0) {
    tdm_load_tile32((unsigned)(uintptr_t)(void*)&lA[0][0],
                    A + (size_t)m0 * K, TM, K);
    tdm_load_tile32((unsigned)(uintptr_t)(void*)&lB[0][0],
                    W + (size_t)n0 * K, TN, K);
  }

  for (int kt = 0; kt < nt; ++kt) {
    const int cur = kt & 1;
    if (wave == 0) {
      if (kt + 1 < nt) {
        tdm_load_tile32((unsigned)(uintptr_t)(void*)&lA[cur ^ 1][0],
                        A + (size_t)m0 * K + (kt + 1) * TK, TM, K);
        tdm_load_tile32((unsigned)(uintptr_t)(void*)&lB[cur ^ 1][0],
                        W + (size_t)n0 * K + (kt + 1) * TK, TN, K);
        __builtin_amdgcn_s_wait_tensorcnt(2);   // tile kt landed, 2 in flight
      } else {
        __builtin_amdgcn_s_wait_tensorcnt(0);   // last tile landed
      }
    }
    __syncthreads();                            // tile kt visible to all waves

#pragma unroll
    for (int kk = 0; kk < TK; kk += 4) {
      const int kb = kk + 2 * lhalf;            // lane-half k base
      v2f a0 = *(const v2f*)(&lA[cur][(wrow +      l16) * SL + kb]);
      v2f a1 = *(const v2f*)(&lA[cur][(wrow + 16 + l16) * SL + kb]);
#pragma unroll
      for (int j = 0; j < 4; ++j) {
        v2f bf = *(const v2f*)(&lB[cur][(j * 16 + l16) * SL + kb]);
        acc[0][j] = wmma4(a0, bf, acc[0][j]);
        acc[1][j] = wmma4(a1, bf, acc[1][j]);
      }
    }
    __syncthreads();       // all waves done with buf[cur] before reload
  }

  // --- epilogue ---
#pragma unroll
  for (int rt = 0; rt < 2; ++rt) {
    const int rowbase = m0 + wrow + rt * 16 + lhalf * 8;  // C layout: v[r]->M=r(+8)
#pragma unroll
    for (int j = 0; j < 4; ++j) {
      const int col = n0 + j * 16 + l16;
      const float bb = bias[col];
      float iv = 0.f, mn = 0.f, bt = 0.f;
      if (mode == 1) {
        iv = gamma[col] * rsqrtf(var[col] + 1e-5f);
        mn = mean[col];
        bt = beta[col];
      }
#pragma unroll
      for (int r = 0; r < 8; ++r) {
        const size_t o = (size_t)(rowbase + r) * N + col;
        float v = acc[rt][j][r] + bb;
        if (mode == 1) v = xres[o] + (v - mn) * iv + bt;
        out[o] = v;
      }
    }
  }
}

// ---------------------------------------------------------------------------
// Attention (per batch b, per 16-row strip s):
//   R_strip(16 x 112) = th[m0:m0+16, :] @ ph^T / 100   (cols >= 100 zeroed)
//   y_strip(16 x 512) = R_strip @ g                    (rows >= 100 dropped)
// One wave per block; R_strip parked in LDS between the two WMMA phases.
// ---------------------------------------------------------------------------
#define SR 120

__global__ __launch_bounds__(32)
void attn_wmma_f32(const float* __restrict__ THm, const float* __restrict__ PHm,
                   const float* __restrict__ Gm, float* __restrict__ Ym) {
  __shared__ float Rs[16 * SR];

  const int b     = blockIdx.x;
  const int strip = blockIdx.y;          // 0..6
  const int lane  = threadIdx.x;
  const int l16   = lane & 15;
  const int lhalf = lane >> 4;
  const int m0    = strip * 16;

  const float* th = THm + (size_t)b * NPTS * CCH;
  const float* ph = PHm + (size_t)b * NPTS * CCH;
  const float* g  = Gm  + (size_t)b * NPTS * CCH;

  // ---- stage 1: R strip over K = 512 ----
  v8f racc[7] = {};
  const int arow = min(m0 + l16, NPTS - 1);   // clamp padded rows
#pragma unroll 2
  for (int k = 0; k < CCH; k += 4) {
    const int kb = k + 2 * lhalf;
    v2f afrag = *(const v2f*)(th + (size_t)arow * CCH + kb);
#pragma unroll
    for (int t = 0; t < 7; ++t) {
      const int brow = min(t * 16 + l16, NPTS - 1);
      v2f bfrag = *(const v2f*)(ph + (size_t)brow * CCH + kb);
      racc[t] = wmma4(afrag, bfrag, racc[t]);
    }
  }

  // scale by 1/N, zero padded columns, park in LDS
#pragma unroll
  for (int t = 0; t < 7; ++t) {
    const int col = t * 16 + l16;
    const float s = (col < NPTS) ? 0.01f : 0.0f;
#pragma unroll
    for (int r = 0; r < 8; ++r) {
      Rs[(r + 8 * lhalf) * SR + col] = racc[t][r] * s;
    }
  }
  __syncthreads();

  // ---- stage 2: y strip = Rs(16x112) @ g(112x512) ----
  for (int j = 0; j < 32; ++j) {          // 32 col tiles of 16
    v8f acc = {};
    for (int kt = 0; kt < 7; ++kt) {
#pragma unroll
      for (int kk = 0; kk < 16; kk += 4) {
        const int kb = kt * 16 + kk + 2 * lhalf;
        v2f afrag = *(const v2f*)(&Rs[l16 * SR + kb]);
        v2f bfrag;
        const int r0 = min(kb, NPTS - 1);       // padded rows hit zeroed R cols
        const int r1 = min(kb + 1, NPTS - 1);
        bfrag.x = g[(size_t)r0 * CCH + j * 16 + l16];
        bfrag.y = g[(size_t)r1 * CCH + j * 16 + l16];
        acc = wmma4(afrag, bfrag, acc);
      }
    }
#pragma unroll
    for (int r = 0; r < 8; ++r) {
      const int row = m0 + r + 8 * lhalf;
      if (row < NPTS)
        Ym[((size_t)b * NPTS + row) * CCH + j * 16 + l16] = acc[r];
    }
  }
}

// ---------------------------------------------------------------------------
extern "C" void kernel_launch(void* const* d_in, const int* in_sizes, int n_in,
                              void* d_out, int out_size, void* d_ws, size_t ws_size,
                              hipStream_t stream) {
  const float* img     = (const float*)d_in[0];
  const float* trans_w = (const float*)d_in[1];
  const float* trans_b = (const float*)d_in[2];
  const float* gw      = (const float*)d_in[3];
  const float* gb      = (const float*)d_in[4];
  const float* tw      = (const float*)d_in[5];
  const float* tb      = (const float*)d_in[6];
  const float* pw      = (const float*)d_in[7];
  const float* pb      = (const float*)d_in[8];
  const float* ww      = (const float*)d_in[9];
  const float* wb      = (const float*)d_in[10];
  const float* bn_g    = (const float*)d_in[11];
  const float* bn_b    = (const float*)d_in[12];
  const float* bn_m    = (const float*)d_in[13];
  const float* bn_v    = (const float*)d_in[14];

  const int M = BDIM * NPTS;          // 25600
  const size_t SZ = (size_t)M * CCH;  // floats per activation buffer

  float* X  = (float*)d_ws;
  float* G  = X  + SZ;
  float* TH = G  + SZ;
  float* PH = TH + SZ;
  float* Y  = PH + SZ;

  const dim3 blk(128);
  const dim3 grd(M / TM, CCH / TN);   // 200 x 8

  // Encoder: X = img @ trans_w^T + trans_b
  gemm_wmma_f32<<<grd, blk, 0, stream>>>(img, trans_w, trans_b, X,
                                         nullptr, nullptr, nullptr, nullptr, nullptr,
                                         M, CCH, DIMG_K, 0);

  for (int l = 0; l < 6; ++l) {
    const size_t wo = (size_t)l * CCH * CCH;
    const size_t bo = (size_t)l * CCH;

    gemm_wmma_f32<<<grd, blk, 0, stream>>>(X, gw + wo, gb + bo, G,
                                           nullptr, nullptr, nullptr, nullptr, nullptr,
                                           M, CCH, CCH, 0);
    gemm_wmma_f32<<<grd, blk, 0, stream>>>(X, tw + wo, tb + bo, TH,
                                           nullptr, nullptr, nullptr, nullptr, nullptr,
                                           M, CCH, CCH, 0);
    gemm_wmma_f32<<<grd, blk, 0, stream>>>(X, pw + wo, pb + bo, PH,
                                           nullptr, nullptr, nullptr, nullptr, nullptr,
                                           M, CCH, CCH, 0);

    attn_wmma_f32<<<dim3(BDIM, NPAD / 16), dim3(32), 0, stream>>>(TH, PH, G, Y);

    // wy GEMM + BN(eval) + residual, in place on X
    gemm_wmma_f32<<<grd, blk, 0, stream>>>(Y, ww + wo, wb + bo, X,
                                           X, bn_g + bo, bn_v + bo, bn_m + bo, bn_b + bo,
                                           M, CCH, CCH, 1);
  }

  (void)hipMemcpyAsync(d_out, X, SZ * sizeof(float), hipMemcpyDeviceToDevice,
                       stream);
}